// RegionWeightedLoss_61100204753358
// MI455X (gfx1250) — compile-verified
//
#include <hip/hip_runtime.h>
#include <hip/hip_bf16.h>

// ---------------------------------------------------------------------------
// RegionWeightedLoss: mean( w(h) * BCEWithLogits(x, t) ) over N=32Mi elements.
// Memory-bound: 256 MiB read @ 23.3 TB/s => ~11.5 us floor. Strategy:
//  - b128 vector loads (float4 logits / int4 indices), grid-stride, 1Mi threads
//  - per-element math in f32 VALU (v_exp_f32 / v_log_f32 hardware trans)
//  - accumulation through V_WMMA_F32_16X16X4_F32 with A = ones (matrix engine
//    performs the reduction adds, full f32 precision, 64 terms/instruction)
//  - deterministic 2-pass reduction: block partials in d_ws -> 1-block finish
// ---------------------------------------------------------------------------

typedef float v2f __attribute__((ext_vector_type(2)));
typedef float v8f __attribute__((ext_vector_type(8)));

#define RWL_NBLOCKS 4096
#define RWL_NTHREADS 256

// D = ones(16x4) x B(4x16) + C : every D row = column sums of B.
__device__ __forceinline__ v8f wmma_ones_acc(v2f b, v8f c) {
    v2f a = {1.0f, 1.0f};
    // 8 args: (neg_a, A, neg_b, B, c_mod, C, reuse_a, reuse_b)
    return __builtin_amdgcn_wmma_f32_16x16x4_f32(
        false, a, false, b, (short)0, c, false, false);
}

__device__ __forceinline__ float wave_reduce_add(float v) {
    // wave32: butterfly over all 32 lanes; every lane ends with the total
    #pragma unroll
    for (int off = 16; off > 0; off >>= 1)
        v += __shfl_xor(v, off, 32);
    return v;
}

__device__ __forceinline__ float bce_weighted(float x, int h, float t, int mid) {
    float a = __builtin_fabsf(x);
    // stable BCEWithLogits: max(x,0) - x*t + log1p(exp(-|x|))
    float l = __logf(1.0f + __expf(-a));
    float b = fmaxf(x, 0.0f) - x * t + l;
    float w = (h >= mid) ? 2.0f : 1.0f;
    return b * w;
}

__global__ void __launch_bounds__(RWL_NTHREADS)
rwl_partials_kernel(const float* __restrict__ xs,
                    const int* __restrict__ hs,
                    const int* __restrict__ tgt,
                    const int* __restrict__ Hp,
                    float* __restrict__ partials,
                    int n) {
    const float t   = (float)tgt[0];
    const int   mid = Hp[0] >> 1;

    const int n4     = n >> 2;
    const int gid    = blockIdx.x * blockDim.x + threadIdx.x;
    const int stride = gridDim.x * blockDim.x;

    const float4* __restrict__ x4 = (const float4*)xs;
    const int4*   __restrict__ h4 = (const int4*)hs;

    v8f c = {0.f, 0.f, 0.f, 0.f, 0.f, 0.f, 0.f, 0.f};

    // Uniform trip count so EXEC is all-1s at every WMMA (only loads predicated).
    const int iters = (n4 + stride - 1) / stride;
    for (int k = 0; k < iters; ++k) {
        const int i = gid + k * stride;
        float4 xv = make_float4(0.f, 0.f, 0.f, 0.f);
        int4   hv = make_int4(0, 0, 0, 0);
        if (i < n4) {
            xv = x4[i];
            hv = h4[i];
            // pull next grid-stride tile toward L2 (global_prefetch_b8)
            __builtin_prefetch((const void*)(x4 + i + stride), 0, 1);
            __builtin_prefetch((const void*)(h4 + i + stride), 0, 1);
        }
        const float s01 = bce_weighted(xv.x, hv.x, t, mid)
                        + bce_weighted(xv.y, hv.y, t, mid);
        const float s23 = bce_weighted(xv.z, hv.z, t, mid)
                        + bce_weighted(xv.w, hv.w, t, mid);
        v2f b = {s01, s23};
        c = wmma_ones_acc(b, c);   // matrix engine folds 64 terms/wave/iter
    }

    // D rows are identical (A == ones): lanes 0-15 hold row M=0, lanes 16-31
    // hold row M=8 of VGPR0 => full-lane sum double-counts; halve it.
    float wsum = 0.5f * wave_reduce_add(c[0]);

    // scalar tail (n % 4), handled once by global thread 0 (lane 0, wave 0)
    if (blockIdx.x == 0 && threadIdx.x == 0) {
        for (int i = (n4 << 2); i < n; ++i)
            wsum += bce_weighted(xs[i], hs[i], t, mid);
    }

    __shared__ float lws[RWL_NTHREADS / 32];
    const int wave = threadIdx.x >> 5;
    const int lane = threadIdx.x & 31;
    if (lane == 0) lws[wave] = wsum;
    __syncthreads();
    if (threadIdx.x == 0) {
        float bsum = 0.f;
        #pragma unroll
        for (int w = 0; w < RWL_NTHREADS / 32; ++w) bsum += lws[w];
        partials[blockIdx.x] = bsum;
    }
}

__global__ void __launch_bounds__(RWL_NTHREADS)
rwl_finalize_kernel(const float* __restrict__ partials,
                    float* __restrict__ out,
                    int nparts, float inv_n) {
    float s = 0.f;
    for (int i = threadIdx.x; i < nparts; i += RWL_NTHREADS)
        s += partials[i];
    s = wave_reduce_add(s);

    __shared__ float lws[RWL_NTHREADS / 32];
    const int wave = threadIdx.x >> 5;
    const int lane = threadIdx.x & 31;
    if (lane == 0) lws[wave] = s;
    __syncthreads();
    if (threadIdx.x == 0) {
        float tot = 0.f;
        #pragma unroll
        for (int w = 0; w < RWL_NTHREADS / 32; ++w) tot += lws[w];
        out[0] = tot * inv_n;
    }
}

extern "C" void kernel_launch(void* const* d_in, const int* in_sizes, int n_in,
                              void* d_out, int out_size, void* d_ws, size_t ws_size,
                              hipStream_t stream) {
    // setup_inputs order: d_out (f32, N), target (int scalar),
    //                     height_indices (i32, N), H (int scalar)
    const float* xs  = (const float*)d_in[0];
    const int*   tgt = (const int*)d_in[1];
    const int*   hs  = (const int*)d_in[2];
    const int*   Hp  = (const int*)d_in[3];
    const int    n   = in_sizes[0];

    float* partials = (float*)d_ws;   // RWL_NBLOCKS floats (16 KiB) of scratch

    rwl_partials_kernel<<<RWL_NBLOCKS, RWL_NTHREADS, 0, stream>>>(
        xs, hs, tgt, Hp, partials, n);
    rwl_finalize_kernel<<<1, RWL_NTHREADS, 0, stream>>>(
        partials, (float*)d_out, RWL_NBLOCKS, 1.0f / (float)n);
}